// RegionGraphReranker_17428977287514
// MI455X (gfx1250) — compile-verified
//
#include <hip/hip_runtime.h>
#include <math.h>

typedef __attribute__((ext_vector_type(2))) float v2f;
typedef __attribute__((ext_vector_type(8))) float v8f;

#define K_PAGES 512
#define T_TOK   784
#define D_DIM   128
#define R_REG   4
#define TN      4
#define BASE_R  (K_PAGES)
#define BASE_T  (K_PAGES + K_PAGES * R_REG)
#define N_NODES (K_PAGES + K_PAGES * R_REG + K_PAGES * TN + 1) // 4609
#define Q_NODE  (N_NODES - 1)
#define F1      256
#define F2      64

__device__ __forceinline__ float4 f4add(float4 a, float4 b) {
  return make_float4(a.x + b.x, a.y + b.y, a.z + b.z, a.w + b.w);
}
__device__ __forceinline__ float4 f4scale(float4 a, float s) {
  return make_float4(a.x * s, a.y * s, a.z * s, a.w * s);
}

// ---------------------------------------------------------------------------
// Pass 1: the only kernel that touches the 205 MB input (roofline ~9us).
// One page per block, 4 waves x 196 tokens, float4 per lane -> each wave
// issues one 512B-coalesced global_load_b128 per token row. Token index is
// wave-uniform so all region/typed membership tests are scalar branches.
// ---------------------------------------------------------------------------
__global__ void reduce_kernel(const float* __restrict__ pm, float* __restrict__ X) {
  const int page = blockIdx.x;
  const int tid  = threadIdx.x;           // 128 = 4 waves
  const int w    = tid >> 5;              // token quarter
  const int lane = tid & 31;              // channel group (4 ch per lane)
  const float4* p = (const float4*)(pm + (size_t)page * T_TOK * D_DIM);

  float4 acc[9];                          // pv, r00,r01,r10,r11, hd,tx,cp,tb
#pragma unroll
  for (int i = 0; i < 9; ++i) acc[i] = make_float4(0.f, 0.f, 0.f, 0.f);

  const int t0 = w * 196, t1 = t0 + 196;
  for (int t = t0; t < t1; ++t) {
    float4 v = p[(size_t)t * 32 + lane];
    const int r = t / 28;
    const int c = t - r * 28;
    acc[0] = f4add(acc[0], v);
    const bool rb = (r >= 14), cb = (c >= 14);       // uniform in wave
    if (!rb && !cb) acc[1] = f4add(acc[1], v);
    if (!rb &&  cb) acc[2] = f4add(acc[2], v);
    if ( rb && !cb) acc[3] = f4add(acc[3], v);
    if ( rb &&  cb) acc[4] = f4add(acc[4], v);
    if (r < 5)                acc[5] = f4add(acc[5], v);   // header [0,5)
    if (r >= 5 && r < 21)     acc[6] = f4add(acc[6], v);   // text   [5,21)
    if (r >= 22)              acc[7] = f4add(acc[7], v);   // caption[22,28)
    if (r >= 9 && r < 19 && c >= 2 && c < 26)
                              acc[8] = f4add(acc[8], v);   // table
  }

  __shared__ float4 sred[128];
  const float scl[9] = {1.f / 784.f, 1.f / 196.f, 1.f / 196.f, 1.f / 196.f,
                        1.f / 196.f, 1.f / 140.f, 1.f / 448.f, 1.f / 168.f,
                        1.f / 240.f};
  const int rowid[9] = {page,
                        BASE_R + page * R_REG + 0, BASE_R + page * R_REG + 1,
                        BASE_R + page * R_REG + 2, BASE_R + page * R_REG + 3,
                        BASE_T + page * TN + 0, BASE_T + page * TN + 1,
                        BASE_T + page * TN + 2, BASE_T + page * TN + 3};
#pragma unroll
  for (int s = 0; s < 9; ++s) {
    sred[tid] = acc[s];
    __syncthreads();
    if (w == 0) {
      float4 v = f4add(f4add(sred[lane], sred[lane + 32]),
                       f4add(sred[lane + 64], sred[lane + 96]));
      ((float4*)X)[(size_t)rowid[s] * 32 + lane] = f4scale(v, scl[s]);
    }
    __syncthreads();
  }
}

__global__ void copy_query(const float* __restrict__ q, float* __restrict__ X) {
  X[(size_t)Q_NODE * D_DIM + threadIdx.x] = q[threadIdx.x];
}

// L2-normalize the 512 page vectors (for page_sim only).
__global__ void norm_pages(const float* __restrict__ X, float* __restrict__ pvn) {
  __shared__ float s[128];
  const int i = blockIdx.x, d = threadIdx.x;
  float v = X[(size_t)i * D_DIM + d];
  s[d] = v * v;
  __syncthreads();
  for (int o = 64; o > 0; o >>= 1) {
    if (d < o) s[d] += s[d + o];
    __syncthreads();
  }
  float n = fmaxf(sqrtf(s[0]), 1e-12f);
  pvn[(size_t)i * D_DIM + d] = v / n;
}

// Cross-page typed edge weights: max(0.7, cos(tv[i,t], tv[i+1,t])).
__global__ void typed_cross(const float* __restrict__ X, float* __restrict__ wx) {
  const int pair = blockIdx.x;            // (K-1)*TN blocks
  const int i = pair / TN, t = pair % TN;
  const int d = threadIdx.x;
  const float* a = X + (size_t)(BASE_T + i * TN + t) * D_DIM;
  const float* b = X + (size_t)(BASE_T + (i + 1) * TN + t) * D_DIM;
  __shared__ float sab[128], saa[128], sbb[128];
  float av = a[d], bv = b[d];
  sab[d] = av * bv; saa[d] = av * av; sbb[d] = bv * bv;
  __syncthreads();
  for (int o = 64; o > 0; o >>= 1) {
    if (d < o) { sab[d] += sab[d + o]; saa[d] += saa[d + o]; sbb[d] += sbb[d + o]; }
    __syncthreads();
  }
  if (d == 0) {
    float na = fmaxf(sqrtf(saa[0]), 1e-12f);
    float nb = fmaxf(sqrtf(sbb[0]), 1e-12f);
    wx[pair] = fmaxf(0.7f, sab[0] / (na * nb));
  }
}

// Naive transpose for tiny L2-resident weight/activation blocks.
__global__ void transpose_kernel(const float* __restrict__ in, float* __restrict__ out,
                                 int rows, int cols) {
  int idx = blockIdx.x * blockDim.x + threadIdx.x;
  if (idx >= rows * cols) return;
  int r = idx / cols, c = idx - r * cols;
  out[(size_t)c * rows + r] = in[idx];
}

// ---------------------------------------------------------------------------
// fp32 GEMM via V_WMMA_F32_16X16X4_F32: C[MxN] = A[MxK] * Bt[NxK]^T.
// BOTH operands K-major -> both fragment loads are contiguous float2
// (global_load_b64), no strided/branchy paths in the k-loop. One 16x16 tile
// per wave32; K accumulated in steps of 4 per the ISA fragment layout
// (A/B: lanes 0-15 hold K=0,1; lanes 16-31 hold K=2,3; C/D: VGPR v -> row
// m0+v (+8 for upper lanes)). Edge rows: clamped loads + guarded stores so
// EXEC stays all-ones through every WMMA.
// ---------------------------------------------------------------------------
__global__ void gemm_wmma_tn(const float* __restrict__ A, const float* __restrict__ Bt,
                             float* __restrict__ C, int M, int N, int Kd) {
  const int wave = threadIdx.x >> 5;
  const int lane = threadIdx.x & 31;
  const int tilesN = N >> 4;
  const int tilesM = (M + 15) >> 4;
  const int tile = blockIdx.x * (blockDim.x >> 5) + wave;
  if (tile >= tilesM * tilesN) return;    // wave-uniform exit
  const int m0 = (tile / tilesN) << 4;
  const int n0 = (tile % tilesN) << 4;
  const int half = lane >> 4;
  const int l16 = lane & 15;
  int arow = m0 + l16; if (arow >= M) arow = M - 1;  // clamped edge loads
  const float* Ar = A + (size_t)arow * Kd + half * 2;
  const float* Br = Bt + (size_t)(n0 + l16) * Kd + half * 2;
  v8f c = {0.f, 0.f, 0.f, 0.f, 0.f, 0.f, 0.f, 0.f};
#pragma unroll 4
  for (int k = 0; k < Kd; k += 4) {
    v2f a = *(const v2f*)(Ar + k);
    v2f b = *(const v2f*)(Br + k);
    c = __builtin_amdgcn_wmma_f32_16x16x4_f32(false, a, false, b, (short)0, c,
                                              false, false);
  }
#pragma unroll
  for (int v = 0; v < 8; ++v) {
    int row = m0 + v + half * 8;
    if (row < M) C[(size_t)row * N + n0 + l16] = c[v];
  }
}

// Page-page adjacency block (only dense part of adj) + page row sums.
__global__ void build_pp(const float* __restrict__ S, const int* __restrict__ pn,
                         float* __restrict__ pp, float* __restrict__ rowsum) {
  const int i = blockIdx.x;
  const int tid = threadIdx.x;            // 256
  __shared__ float red[256];
  float local = 0.f;
  const int pni = pn[i];
  for (int j = tid; j < K_PAGES; j += 256) {
    float s = S[(size_t)i * K_PAGES + j];
    float p = (j != i && s >= 0.7f) ? s : 0.f;
    int dpn = pn[j] - pni; if (dpn < 0) dpn = -dpn;
    if (dpn == 1) p = fmaxf(p, 0.8f);
    pp[(size_t)i * K_PAGES + j] = p;
    local += p;
  }
  red[tid] = local; __syncthreads();
  for (int o = 128; o > 0; o >>= 1) {
    if (tid < o) red[tid] += red[tid + o];
    __syncthreads();
  }
  // + regions(4) + typed(4) + query(1) + self(1)
  if (tid == 0) rowsum[i] = red[0] + 10.f;
}

__global__ void zero_f32(float* __restrict__ p, int n) {
  int i = blockIdx.x * blockDim.x + threadIdx.x;
  if (i < n) p[i] = 0.f;
}

// Column sums of G (for the query row of adj @ G). blockDim = F.
__global__ void colsum_kernel(const float* __restrict__ G, float* __restrict__ cs,
                              int rows, int F) {
  const int f = threadIdx.x;
  int r0 = blockIdx.x * 128;
  int r1 = r0 + 128; if (r1 > rows) r1 = rows;
  float acc = 0.f;
  for (int r = r0; r < r1; ++r) acc += G[(size_t)r * F + f];
  atomicAdd(&cs[f], acc);
}

// H1 = elu(adj @ G1), exploiting adjacency structure (row type is
// block-uniform, so no divergence). blockDim = F1.
__global__ void combine1(const float* __restrict__ G, const float* __restrict__ PPG,
                         const float* __restrict__ cs, const float* __restrict__ rsp,
                         const float* __restrict__ wx, float* __restrict__ H) {
  const int row = blockIdx.x;
  const int f = threadIdx.x;
  const int F = F1;
  const float gq = G[(size_t)Q_NODE * F + f];
  float acc, rs;
  if (row < K_PAGES) {
    const int i = row;
    acc = PPG[(size_t)i * F + f] + gq + G[(size_t)i * F + f];
    for (int a = 0; a < R_REG; ++a) acc += G[(size_t)(BASE_R + i * R_REG + a) * F + f];
    for (int t = 0; t < TN; ++t)    acc += G[(size_t)(BASE_T + i * TN + t) * F + f];
    rs = rsp[i];
  } else if (row < BASE_T) {
    const int i = (row - BASE_R) >> 2, a = (row - BASE_R) & 3;
    const float* Gr = G + (size_t)(BASE_R + i * R_REG) * F + f;
    acc = G[(size_t)i * F + f] + gq + Gr[(size_t)a * F]
        + Gr[(size_t)(a ^ 1) * F] + Gr[(size_t)(a ^ 2) * F]
        + 0.9f * Gr[(size_t)(a ^ 3) * F];
    rs = 5.9f;                            // 1+1+1 + 1+1+0.9
  } else if (row < Q_NODE) {
    const int i = (row - BASE_T) >> 2, t = (row - BASE_T) & 3;
    const float* Gt = G + (size_t)BASE_T * F + f;
    const float wl = (i > 0) ? wx[(i - 1) * TN + t] : 0.f;
    const float wr = (i < K_PAGES - 1) ? wx[i * TN + t] : 0.f;
    acc = G[(size_t)i * F + f] + gq
        + Gt[(size_t)(i * TN + t) * F]          // self
        + Gt[(size_t)(i * TN + (t ^ 1)) * F];   // intra partner
    if (i > 0)           acc += wl * Gt[(size_t)((i - 1) * TN + t) * F];
    if (i < K_PAGES - 1) acc += wr * Gt[(size_t)((i + 1) * TN + t) * F];
    rs = 4.f + wl + wr;
  } else {
    acc = cs[f];                          // colsum includes G1[q] once = self
    rs = (float)N_NODES;
  }
  const float x = acc / rs;
  H[(size_t)row * F + f] = (x > 0.f) ? x : (expf(x) - 1.f);
}

// Second adj product only needed for page rows; fuse head + blend.
__global__ void final_kernel(const float* __restrict__ G2, const float* __restrict__ PPG2,
                             const float* __restrict__ rsp, const float* __restrict__ wh,
                             const float* __restrict__ init, float* __restrict__ out) {
  const int i = blockIdx.x;
  const int f = threadIdx.x;              // 64
  __shared__ float red[64];
  const int F = F2;
  float acc = PPG2[(size_t)i * F + f] + G2[(size_t)Q_NODE * F + f]
            + G2[(size_t)i * F + f];
  for (int a = 0; a < R_REG; ++a) acc += G2[(size_t)(BASE_R + i * R_REG + a) * F + f];
  for (int t = 0; t < TN; ++t)    acc += G2[(size_t)(BASE_T + i * TN + t) * F + f];
  red[f] = acc / rsp[i] * wh[f];
  __syncthreads();
  for (int o = 32; o > 0; o >>= 1) {
    if (f < o) red[f] += red[f + o];
    __syncthreads();
  }
  if (f == 0) out[i] = 0.5f * init[i] + 0.5f * red[0];
}

extern "C" void kernel_launch(void* const* d_in, const int* in_sizes, int n_in,
                              void* d_out, int out_size, void* d_ws, size_t ws_size,
                              hipStream_t stream) {
  (void)in_sizes; (void)n_in; (void)out_size; (void)ws_size;
  const float* pm   = (const float*)d_in[0];
  const float* qv   = (const float*)d_in[1];
  const float* init = (const float*)d_in[2];
  const float* W1   = (const float*)d_in[3];
  const float* W2   = (const float*)d_in[4];
  const float* wh   = (const float*)d_in[5];
  const int*   pn   = (const int*)d_in[6];
  float* out = (float*)d_out;

  // Workspace carve-up (floats). Total ~17 MB.
  float* ws = (float*)d_ws;
  size_t off = 0;
  auto alloc = [&](size_t n) { float* p = ws + off; off += (n + 255) & ~(size_t)255; return p; };
  float* X    = alloc((size_t)N_NODES * D_DIM);   // node features (K-major)
  float* pvn  = alloc((size_t)K_PAGES * D_DIM);   // normalized page vecs
  float* S    = alloc((size_t)K_PAGES * K_PAGES); // page_sim
  float* pp   = alloc((size_t)K_PAGES * K_PAGES); // page-page adj block
  float* rsp  = alloc(K_PAGES);                   // page row sums
  float* wx   = alloc((size_t)(K_PAGES - 1) * TN);// cross typed weights
  float* Wt1  = alloc((size_t)F1 * D_DIM);        // W1^T (256x128)
  float* Wt2  = alloc((size_t)F2 * F1);           // W2^T (64x256)
  float* G1   = alloc((size_t)N_NODES * F1);      // X @ W1
  float* G1tp = alloc((size_t)F1 * K_PAGES);      // (G1[:K])^T (256x512)
  float* PPG1 = alloc((size_t)K_PAGES * F1);      // pp @ G1_pages
  float* cs1  = alloc(F1);                        // column sums of G1
  float* H1   = alloc((size_t)N_NODES * F1);      // elu(adj @ G1)
  float* G2   = alloc((size_t)N_NODES * F2);      // H1 @ W2
  float* G2tp = alloc((size_t)F2 * K_PAGES);      // (G2[:K])^T (64x512)
  float* PPG2 = alloc((size_t)K_PAGES * F2);      // pp @ G2_pages

  auto gemm = [&](const float* A, const float* Bt, float* C, int M, int N, int Kd) {
    int tiles = ((M + 15) >> 4) * (N >> 4);
    int grid = (tiles + 3) / 4;                   // 4 waves (128 thr) per block
    gemm_wmma_tn<<<grid, 128, 0, stream>>>(A, Bt, C, M, N, Kd);
  };
  auto transp = [&](const float* in, float* o, int rows, int cols) {
    int n = rows * cols;
    transpose_kernel<<<(n + 255) / 256, 256, 0, stream>>>(in, o, rows, cols);
  };

  reduce_kernel<<<K_PAGES, 128, 0, stream>>>(pm, X);
  copy_query<<<1, 128, 0, stream>>>(qv, X);
  norm_pages<<<K_PAGES, 128, 0, stream>>>(X, pvn);
  typed_cross<<<(K_PAGES - 1) * TN, 128, 0, stream>>>(X, wx);

  gemm(pvn, pvn, S, K_PAGES, K_PAGES, D_DIM);     // page_sim = pvn @ pvn^T
  build_pp<<<K_PAGES, 256, 0, stream>>>(S, pn, pp, rsp);

  transp(W1, Wt1, D_DIM, F1);                     // 128x256 -> 256x128
  gemm(X, Wt1, G1, N_NODES, F1, D_DIM);           // G1 = X @ W1
  zero_f32<<<1, F1, 0, stream>>>(cs1, F1);
  colsum_kernel<<<(N_NODES + 127) / 128, F1, 0, stream>>>(G1, cs1, N_NODES, F1);
  transp(G1, G1tp, K_PAGES, F1);                  // (G1[:K])^T
  gemm(pp, G1tp, PPG1, K_PAGES, F1, K_PAGES);     // pp @ G1[:K]
  combine1<<<N_NODES, F1, 0, stream>>>(G1, PPG1, cs1, rsp, wx, H1);

  transp(W2, Wt2, F1, F2);                        // 256x64 -> 64x256
  gemm(H1, Wt2, G2, N_NODES, F2, F1);             // G2 = H1 @ W2
  transp(G2, G2tp, K_PAGES, F2);                  // (G2[:K])^T
  gemm(pp, G2tp, PPG2, K_PAGES, F2, K_PAGES);     // pp @ G2[:K]
  final_kernel<<<K_PAGES, F2, 0, stream>>>(G2, PPG2, rsp, wh, init, out);
}